// M11_70480413327934
// MI455X (gfx1250) — compile-verified
//
#include <hip/hip_runtime.h>

typedef __attribute__((ext_vector_type(2))) float v2f;
typedef __attribute__((ext_vector_type(8))) float v8f;

#define NNODES 50000
#define NEDGES 800000
#define FNODE  64
#define FEDGE  16
#define HLD    448      // final concat width: 64 + 3*128
#define NEGS   0.01f
#define BNEPS  1e-5f

// ---------------- copy x into h[:, 0:64] (ld = 448) ----------------
__global__ void copy_x_kernel(const float* __restrict__ x, float* __restrict__ h) {
    int idx = blockIdx.x * 256 + threadIdx.x;
    if (idx >= NNODES * FNODE) return;
    int n = idx / FNODE, c = idx % FNODE;
    h[(size_t)n * HLD + c] = x[idx];
}

// ---------------- per-column mean / rsqrt(var+eps) ----------------
__global__ void colstats_kernel(const float* __restrict__ src, int rows, int ld,
                                float* __restrict__ mu, float* __restrict__ rstd) {
    int c = blockIdx.x;
    __shared__ double ssum[256], ssq[256];
    double s = 0.0, q = 0.0;
    for (int r = threadIdx.x; r < rows; r += 256) {
        float v = src[(size_t)r * ld + c];
        s += v; q += (double)v * v;
    }
    ssum[threadIdx.x] = s; ssq[threadIdx.x] = q;
    __syncthreads();
    for (int off = 128; off > 0; off >>= 1) {
        if (threadIdx.x < off) {
            ssum[threadIdx.x] += ssum[threadIdx.x + off];
            ssq[threadIdx.x]  += ssq[threadIdx.x + off];
        }
        __syncthreads();
    }
    if (threadIdx.x == 0) {
        double m   = ssum[0] / rows;
        double var = ssq[0] / rows - m * m;
        mu[c]   = (float)m;
        rstd[c] = rsqrtf((float)var + BNEPS);
    }
}

// ---------------- y = (x-mu)*rstd*g + b, optional LeakyReLU ----------------
__global__ void bn_kernel(const float* __restrict__ src, float* __restrict__ dst,
                          int rows, int cols, int lds, int ldd,
                          const float* __restrict__ mu, const float* __restrict__ rstd,
                          const float* __restrict__ g, const float* __restrict__ b,
                          int do_lrelu) {
    int idx = blockIdx.x * 256 + threadIdx.x;
    if (idx >= rows * cols) return;
    int r = idx / cols, c = idx % cols;
    float v = (src[(size_t)r * lds + c] - mu[c]) * rstd[c] * g[c] + b[c];
    if (do_lrelu && v < 0.f) v *= NEGS;
    dst[(size_t)r * ldd + c] = v;
}

// ---------------- edge phase: e = ea@ew+eb; msg = relu(hn[src]+e); agg[dst]+=msg --
__global__ void edge_kernel(const float* __restrict__ hn, float* __restrict__ agg,
                            const float* __restrict__ ea,
                            const long long* __restrict__ srcI,
                            const long long* __restrict__ dstI,
                            const float* __restrict__ ew, const float* __restrict__ eb,
                            int ci) {
    extern __shared__ float sh[];               // [16*ci] ew + [ci] eb
    int tid = threadIdx.y * 32 + threadIdx.x;
    for (int i = tid; i < 17 * ci; i += 256)
        sh[i] = (i < 16 * ci) ? ew[i] : eb[i - 16 * ci];
    __syncthreads();
    int e = blockIdx.x * 8 + threadIdx.y;
    if (e >= NEDGES) return;
    int s = (int)srcI[e];
    int d = (int)dstI[e];
    float av[FEDGE];
#pragma unroll
    for (int k = 0; k < FEDGE; k++) av[k] = ea[(size_t)e * FEDGE + k];
    const float* hrow = hn  + (size_t)s * ci;
    float*       arow = agg + (size_t)d * ci;
    for (int c = threadIdx.x; c < ci; c += 32) {
        float acc = sh[16 * ci + c];
#pragma unroll
        for (int k = 0; k < FEDGE; k++) acc += av[k] * sh[k * ci + c];
        float m = hrow[c] + acc;
        m = m > 0.f ? m : 0.f;                  // ReLU
        __hip_atomic_fetch_add(&arow[c], m, __ATOMIC_RELAXED, __HIP_MEMORY_SCOPE_AGENT);
    }
}

// ---------------- hn = (1+eps)*hn + agg ----------------
__global__ void hpre_kernel(float* __restrict__ hn, const float* __restrict__ agg,
                            const float* __restrict__ epsp, int total) {
    int idx = blockIdx.x * 256 + threadIdx.x;
    if (idx >= total) return;
    hn[idx] = (1.f + epsp[0]) * hn[idx] + agg[idx];
}

// ---------------- fp32 WMMA GEMM: D[m,n] = sum_k A[m,k]*W[k,n] + bias[n] -------
// One wave per 16x16 tile. W is row-major [K x 128].
// A operand layout (V_WMMA_F32_16X16X4_F32): lane l = 16*h + ln holds
//   v0 = A[ln][2h], v1 = A[ln][2h+1];  B mirrored: v0 = B[2h][ln], v1 = B[2h+1][ln].
__global__ void wmma_gemm_kernel(const float* __restrict__ A, int lda, int K,
                                 const float* __restrict__ W,
                                 const float* __restrict__ bias,
                                 float* __restrict__ D, int ldd) {
    int m0 = blockIdx.x * 16;
    int n0 = blockIdx.y * 16;
    int l  = threadIdx.x;
    int hh = l >> 4;
    int ln = l & 15;
    v8f acc = {};
    const float* arow = A + (size_t)(m0 + ln) * lda + 2 * hh;
    const float* wcol = W + n0 + ln;
    for (int k0 = 0; k0 < K; k0 += 4) {
        v2f a, b;
        a.x = arow[k0];
        a.y = arow[k0 + 1];
        b.x = wcol[(size_t)(k0 + 2 * hh) * 128];
        b.y = wcol[(size_t)(k0 + 2 * hh + 1) * 128];
        acc = __builtin_amdgcn_wmma_f32_16x16x4_f32(false, a, false, b,
                                                    (short)0, acc, false, false);
    }
    float bb = bias[n0 + ln];
#pragma unroll
    for (int r = 0; r < 8; r++)
        D[(size_t)(m0 + r + 8 * hh) * ldd + n0 + ln] = acc[r] + bb;
}

// ---------------- final matvec: out[n] = t[n,:] . w2 + b2 ----------------
__global__ void matvec_kernel(const float* __restrict__ t, const float* __restrict__ w,
                              const float* __restrict__ b, float* __restrict__ out) {
    int n = blockIdx.x * 256 + threadIdx.x;
    if (n >= NNODES) return;
    float acc = b[0];
    const float* row = t + (size_t)n * 128;
#pragma unroll 8
    for (int c = 0; c < 128; c++) acc += row[c] * w[c];
    out[n] = acc;
}

// ================= host side =================
struct LayerP { const float *bn_g,*bn_b,*eps,*ew,*eb,*w1,*b1,*g2,*b2,*w2,*b2b; };
struct ClfP   { const float *w1,*b1,*g,*b,*w2,*b2; };

static inline const float* F(void* const* d, int i) { return (const float*)d[i]; }

extern "C" void kernel_launch(void* const* d_in, const int* in_sizes, int n_in,
                              void* d_out, int out_size, void* d_ws, size_t ws_size,
                              hipStream_t stream) {
    (void)out_size; (void)ws_size;
    const float* x; const long long* ei; const float* ea;
    LayerP LP[3]; ClfP CF;
    int pidx; bool insertion_all = false;

    if (in_sizes[0] == NNODES * FNODE) {              // top-level insertion order
        x  = F(d_in, 0);
        ei = (const long long*)d_in[1];
        ea = F(d_in, 2);
        pidx = 3;
        insertion_all = (in_sizes[3] == FNODE);       // layer0.bn_g first => raw insertion
    } else {                                          // jax-sorted: edge_attr, edge_index, params..., x
        ea = F(d_in, 0);
        ei = (const long long*)d_in[1];
        x  = F(d_in, n_in - 1);
        pidx = 2;
    }

    if (insertion_all) {                              // layers (insertion keys) then clf
        int i2 = pidx;
        for (int i = 0; i < 3; i++) {
            LayerP& p = LP[i];
            p.bn_g = F(d_in, i2++); p.bn_b = F(d_in, i2++); p.eps = F(d_in, i2++);
            p.ew   = F(d_in, i2++); p.eb   = F(d_in, i2++); p.w1  = F(d_in, i2++);
            p.b1   = F(d_in, i2++); p.g2   = F(d_in, i2++); p.b2  = F(d_in, i2++);
            p.w2   = F(d_in, i2++); p.b2b  = F(d_in, i2++);
        }
        CF.w1 = F(d_in, i2++); CF.b1 = F(d_in, i2++); CF.g = F(d_in, i2++);
        CF.b  = F(d_in, i2++); CF.w2 = F(d_in, i2++); CF.b2 = F(d_in, i2++);
    } else {                                          // sorted dict keys: clf then layers
        int i2 = pidx;
        CF.b  = F(d_in, i2++); CF.b1 = F(d_in, i2++); CF.b2 = F(d_in, i2++);
        CF.g  = F(d_in, i2++); CF.w1 = F(d_in, i2++); CF.w2 = F(d_in, i2++);
        for (int i = 0; i < 3; i++) {
            LayerP& p = LP[i];
            p.b1   = F(d_in, i2++); p.b2   = F(d_in, i2++); p.b2b = F(d_in, i2++);
            p.bn_b = F(d_in, i2++); p.bn_g = F(d_in, i2++); p.eb  = F(d_in, i2++);
            p.eps  = F(d_in, i2++); p.ew   = F(d_in, i2++); p.g2  = F(d_in, i2++);
            p.w1   = F(d_in, i2++); p.w2   = F(d_in, i2++);
        }
    }

    const long long* srcI = ei;
    const long long* dstI = ei + NEDGES;

    // workspace layout (floats)
    float* ws   = (float*)d_ws;
    float* h    = ws;                               // N x 448
    float* hn   = h   + (size_t)NNODES * HLD;       // N x 320 (max)
    float* agg  = hn  + (size_t)NNODES * 320;       // N x 320 (max)
    float* t    = agg + (size_t)NNODES * 320;       // N x 128
    float* mu   = t   + (size_t)NNODES * 128;       // 512
    float* rstd = mu  + 512;                        // 512

    copy_x_kernel<<<(NNODES * FNODE + 255) / 256, 256, 0, stream>>>(x, h);

    for (int i = 0; i < 3; i++) {
        int ci = FNODE + 128 * i;                   // 64, 192, 320
        // BN over current concat width (+ LeakyReLU for i>0)
        colstats_kernel<<<ci, 256, 0, stream>>>(h, NNODES, HLD, mu, rstd);
        bn_kernel<<<(NNODES * ci + 255) / 256, 256, 0, stream>>>(
            h, hn, NNODES, ci, HLD, ci, mu, rstd, LP[i].bn_g, LP[i].bn_b, i > 0);
        // edge scatter
        hipMemsetAsync(agg, 0, (size_t)NNODES * ci * sizeof(float), stream);
        edge_kernel<<<(NEDGES + 7) / 8, dim3(32, 8), 17 * ci * sizeof(float), stream>>>(
            hn, agg, ea, srcI, dstI, LP[i].ew, LP[i].eb, ci);
        // h' = (1+eps)*hn + agg, then MLP
        hpre_kernel<<<(NNODES * ci + 255) / 256, 256, 0, stream>>>(hn, agg, LP[i].eps, NNODES * ci);
        wmma_gemm_kernel<<<dim3(NNODES / 16, 8), 32, 0, stream>>>(
            hn, ci, ci, LP[i].w1, LP[i].b1, t, 128);
        colstats_kernel<<<128, 256, 0, stream>>>(t, NNODES, 128, mu, rstd);
        bn_kernel<<<(NNODES * 128 + 255) / 256, 256, 0, stream>>>(
            t, t, NNODES, 128, 128, 128, mu, rstd, LP[i].g2, LP[i].b2, 1);
        wmma_gemm_kernel<<<dim3(NNODES / 16, 8), 32, 0, stream>>>(
            t, 128, 128, LP[i].w2, LP[i].b2b, h + FNODE + 128 * i, HLD);
    }

    // classifier
    wmma_gemm_kernel<<<dim3(NNODES / 16, 8), 32, 0, stream>>>(
        h, HLD, HLD, CF.w1, CF.b1, t, 128);
    colstats_kernel<<<128, 256, 0, stream>>>(t, NNODES, 128, mu, rstd);
    bn_kernel<<<(NNODES * 128 + 255) / 256, 256, 0, stream>>>(
        t, t, NNODES, 128, 128, 128, mu, rstd, CF.g, CF.b, 1);
    matvec_kernel<<<(NNODES + 255) / 256, 256, 0, stream>>>(t, CF.w2, CF.b2, (float*)d_out);
}